// SignNetLayer_Transformer2_88252987998303
// MI455X (gfx1250) — compile-verified
//
#include <hip/hip_runtime.h>
#include <math.h>

#define N_NODES 50000
#define N_EDGES 800000
#define PE 16
#define ED 16
#define OUT 64
#define HID 128
#define NHEAD 8
#define DHEAD 8

typedef float v2f __attribute__((ext_vector_type(2)));
typedef float v8f __attribute__((ext_vector_type(8)));

// D(16x16 f32) = A(16x4 f32) * B(4x16 f32) + C, wave32.
__device__ __forceinline__ v8f wmma_f32_k4(v2f a, v2f b, v8f c) {
  return __builtin_amdgcn_wmma_f32_16x16x4_f32(false, a, false, b, (short)0, c,
                                               false, false);
}

__global__ void zero_kernel(float* __restrict__ p, long n) {
  long i = (long)blockIdx.x * blockDim.x + threadIdx.x;
  long stride = (long)gridDim.x * blockDim.x;
  for (; i < n; i += stride) p[i] = 0.0f;
}

// ---------------------------------------------------------------------------
// Kernel 1: bias-free node projections Qx/Kx/Vx = pos_enc @ W{q,k,v}
// block = 128 threads (4 waves); wave w owns 16-column tile w; 16 nodes/block.
// ---------------------------------------------------------------------------
__global__ void __launch_bounds__(128)
node_proj_kernel(const float* __restrict__ x, const float* __restrict__ Wq,
                 const float* __restrict__ Wk, const float* __restrict__ Wv,
                 float* __restrict__ Qx, float* __restrict__ Kx,
                 float* __restrict__ Vx) {
  const int tile = blockIdx.x;
  const int wave = threadIdx.x >> 5;
  const int lane = threadIdx.x & 31;
  const int row = lane & 15;
  const int khalf = (lane >> 4) << 1;  // 0 or 2
  const int mbase = (lane >> 4) << 3;  // 0 or 8
  const int node = tile * 16 + row;
  const int col = wave * 16 + row;

  // A tile: pos_enc rows, K=16 -> 4 WMMA steps, ISA 16x4-f32 layout.
  v2f a[4];
#pragma unroll
  for (int s = 0; s < 4; ++s) {
    const float* xr = x + (long)node * PE + s * 4 + khalf;
    a[s].x = xr[0];
    a[s].y = xr[1];
  }
  const float* W[3] = {Wq, Wk, Wv};
  float* O[3] = {Qx, Kx, Vx};
#pragma unroll
  for (int m = 0; m < 3; ++m) {
    v8f acc = {};
#pragma unroll
    for (int s = 0; s < 4; ++s) {
      v2f b;
      const float* wp = W[m] + (s * 4 + khalf) * OUT + col;
      b.x = wp[0];
      b.y = wp[OUT];
      acc = wmma_f32_k4(a[s], b, acc);
    }
#pragma unroll
    for (int r = 0; r < 8; ++r)
      O[m][(long)(tile * 16 + mbase + r) * OUT + col] = acc[r];
  }
}

// ---------------------------------------------------------------------------
// Kernel 2: fused edge projection (WMMA -> LDS) + dual-sign attention scatter.
// block = 128 threads (4 waves) handles 16 edges.
// ---------------------------------------------------------------------------
#define EE_STRIDE 68  // 64 + 4 pad: avoids LDS bank conflicts on WMMA stores

__global__ void __launch_bounds__(128)
edge_kernel(const float* __restrict__ eattr, const int* __restrict__ eidx,
            const float* __restrict__ We, const float* __restrict__ be,
            const float* __restrict__ Qx, const float* __restrict__ Kx,
            const float* __restrict__ Vx, const float* __restrict__ bq,
            const float* __restrict__ bk, const float* __restrict__ bv,
            float* __restrict__ num1, float* __restrict__ den1,
            float* __restrict__ num2, float* __restrict__ den2) {
  __shared__ float ee[16 * EE_STRIDE];
  const int tile = blockIdx.x;
  const int wave = threadIdx.x >> 5;
  const int lane = threadIdx.x & 31;
  const int row = lane & 15;
  const int khalf = (lane >> 4) << 1;
  const int mbase = (lane >> 4) << 3;
  const int col = wave * 16 + row;
  const int edge0 = tile * 16;

  // --- Ee tile = edge_attr[16x16] @ We[16x64] + be, via 4 WMMAs per wave ---
  v8f acc = {};
  {
    const float bias = be[col];
#pragma unroll
    for (int r = 0; r < 8; ++r) acc[r] = bias;
  }
#pragma unroll
  for (int s = 0; s < 4; ++s) {
    v2f a, b;
    const float* ar = eattr + (long)(edge0 + row) * ED + s * 4 + khalf;
    a.x = ar[0];
    a.y = ar[1];
    const float* wp = We + (s * 4 + khalf) * OUT + col;
    b.x = wp[0];
    b.y = wp[OUT];
    acc = wmma_f32_k4(a, b, acc);
  }
#pragma unroll
  for (int r = 0; r < 8; ++r) ee[(mbase + r) * EE_STRIDE + col] = acc[r];
  __syncthreads();

  // --- scatter: thread t -> (edge j, head h); both sign branches at once ---
  const int j = threadIdx.x >> 3;
  const int h = threadIdx.x & 7;
  const int e = edge0 + j;
  const int src = eidx[e];
  const int dst = eidx[N_EDGES + e];
  const int hd = h * DHEAD;
  const float* qp = Qx + (long)dst * OUT + hd;
  const float* kp = Kx + (long)src * OUT + hd;
  const float* vp = Vx + (long)src * OUT + hd;
  const float* eep = ee + j * EE_STRIDE + hd;

  float s1 = 0.0f, s2 = 0.0f;
#pragma unroll
  for (int d = 0; d < DHEAD; ++d) {
    const float q = qp[d], k = kp[d], ev = eep[d];
    const float bqv = bq[hd + d], bkv = bk[hd + d];
    s1 += (k + bkv) * (q + bqv) * ev;   // phi(+x)
    s2 += (bkv - k) * (bqv - q) * ev;   // phi(-x)
  }
  const float scale = 0.35355339059327373f;  // 1/sqrt(8)
  const float w1 = __expf(fminf(fmaxf(s1 * scale, -5.0f), 5.0f));
  const float w2 = __expf(fminf(fmaxf(s2 * scale, -5.0f), 5.0f));

  float* n1 = num1 + (long)dst * OUT + hd;
  float* n2 = num2 + (long)dst * OUT + hd;
#pragma unroll
  for (int d = 0; d < DHEAD; ++d) {
    const float v = vp[d], bvv = bv[hd + d];
    atomicAdd(&n1[d], w1 * (v + bvv));
    atomicAdd(&n2[d], w2 * (bvv - v));
  }
  atomicAdd(&den1[(long)dst * NHEAD + h], w1);
  atomicAdd(&den2[(long)dst * NHEAD + h], w2);
}

// ---------------------------------------------------------------------------
// Kernel 3: h = num1/den1 + num2/den2 ; out = relu(h@W1+b1)@W2+b2 (all WMMA)
// block = 128 threads (4 waves) handles 16 nodes.
// ---------------------------------------------------------------------------
#define H_STRIDE 68
#define HID_STRIDE 132

__global__ void __launch_bounds__(128)
mlp_kernel(const float* __restrict__ num1, const float* __restrict__ den1,
           const float* __restrict__ num2, const float* __restrict__ den2,
           const float* __restrict__ W1, const float* __restrict__ b1,
           const float* __restrict__ W2, const float* __restrict__ b2,
           float* __restrict__ out) {
  __shared__ float hbuf[16 * H_STRIDE];
  __shared__ float hid[16 * HID_STRIDE];
  const int tile = blockIdx.x;
  const int wave = threadIdx.x >> 5;
  const int lane = threadIdx.x & 31;
  const int row = lane & 15;
  const int khalf = (lane >> 4) << 1;
  const int mbase = (lane >> 4) << 3;

  // phase 1: finalize h into LDS (8 elements per thread)
  {
    const int j = threadIdx.x >> 3;
    const int c0 = (threadIdx.x & 7) * 8;
    const long n = (long)tile * 16 + j;
    const float d1 = den1[n * NHEAD + (c0 >> 3)] + 1e-6f;
    const float d2 = den2[n * NHEAD + (c0 >> 3)] + 1e-6f;
    const float r1 = 1.0f / d1, r2 = 1.0f / d2;
#pragma unroll
    for (int d = 0; d < 8; ++d) {
      const int c = c0 + d;
      hbuf[j * H_STRIDE + c] = num1[n * OUT + c] * r1 + num2[n * OUT + c] * r2;
    }
  }
  __syncthreads();

  // phase 2: hidden = relu(h @ W1 + b1); 8 col tiles, 2 per wave; K=64
#pragma unroll
  for (int t = 0; t < 2; ++t) {
    const int ct = wave + t * 4;
    const int col = ct * 16 + row;
    v8f acc = {};
    {
      const float bias = b1[col];
#pragma unroll
      for (int r = 0; r < 8; ++r) acc[r] = bias;
    }
#pragma unroll
    for (int s = 0; s < 16; ++s) {
      v2f a, b;
      const int k = s * 4 + khalf;
      a.x = hbuf[row * H_STRIDE + k];
      a.y = hbuf[row * H_STRIDE + k + 1];
      b.x = W1[k * HID + col];
      b.y = W1[(k + 1) * HID + col];
      acc = wmma_f32_k4(a, b, acc);
    }
#pragma unroll
    for (int r = 0; r < 8; ++r)
      hid[(mbase + r) * HID_STRIDE + col] = fmaxf(acc[r], 0.0f);
  }
  __syncthreads();

  // phase 3: out = hidden @ W2 + b2; 4 col tiles, 1 per wave; K=128
  {
    const int col = wave * 16 + row;
    v8f acc = {};
    {
      const float bias = b2[col];
#pragma unroll
      for (int r = 0; r < 8; ++r) acc[r] = bias;
    }
#pragma unroll
    for (int s = 0; s < 32; ++s) {
      v2f a, b;
      const int k = s * 4 + khalf;
      a.x = hid[row * HID_STRIDE + k];
      a.y = hid[row * HID_STRIDE + k + 1];
      b.x = W2[k * OUT + col];
      b.y = W2[(k + 1) * OUT + col];
      acc = wmma_f32_k4(a, b, acc);
    }
#pragma unroll
    for (int r = 0; r < 8; ++r)
      out[(long)(tile * 16 + mbase + r) * OUT + col] = acc[r];
  }
}

extern "C" void kernel_launch(void* const* d_in, const int* in_sizes, int n_in,
                              void* d_out, int out_size, void* d_ws,
                              size_t ws_size, hipStream_t stream) {
  (void)in_sizes; (void)n_in; (void)out_size; (void)ws_size;
  const float* pos_enc = (const float*)d_in[0];
  const float* eattr = (const float*)d_in[1];
  const int* eidx = (const int*)d_in[2];
  const float* Wq = (const float*)d_in[3];
  const float* bq = (const float*)d_in[4];
  const float* Wk = (const float*)d_in[5];
  const float* bk = (const float*)d_in[6];
  const float* Wv = (const float*)d_in[7];
  const float* bv = (const float*)d_in[8];
  const float* We = (const float*)d_in[9];
  const float* be = (const float*)d_in[10];
  const float* W1 = (const float*)d_in[11];
  const float* b1 = (const float*)d_in[12];
  const float* W2 = (const float*)d_in[13];
  const float* b2 = (const float*)d_in[14];
  float* out = (float*)d_out;

  // workspace layout (floats): Qx,Kx,Vx [N,64] ; num1[N,64] den1[N,8] ;
  // num2[N,64] den2[N,8]  => 336*N floats ~= 67 MB
  float* ws = (float*)d_ws;
  float* Qx = ws;
  float* Kx = Qx + (long)N_NODES * OUT;
  float* Vx = Kx + (long)N_NODES * OUT;
  float* num1 = Vx + (long)N_NODES * OUT;
  float* den1 = num1 + (long)N_NODES * OUT;
  float* num2 = den1 + (long)N_NODES * NHEAD;
  float* den2 = num2 + (long)N_NODES * OUT;

  const long zelems = (long)N_NODES * (2 * OUT + 2 * NHEAD);
  zero_kernel<<<1024, 256, 0, stream>>>(num1, zelems);
  node_proj_kernel<<<N_NODES / 16, 128, 0, stream>>>(pos_enc, Wq, Wk, Wv, Qx,
                                                     Kx, Vx);
  edge_kernel<<<N_EDGES / 16, 128, 0, stream>>>(eattr, eidx, We, be, Qx, Kx,
                                                Vx, bq, bk, bv, num1, den1,
                                                num2, den2);
  mlp_kernel<<<N_NODES / 16, 128, 0, stream>>>(num1, den1, num2, den2, W1, b1,
                                               W2, b2, out);
}